// _JetMoARouter_85787676770833
// MI455X (gfx1250) — compile-verified
//
#include <hip/hip_runtime.h>

typedef float v2f __attribute__((ext_vector_type(2)));
typedef float v8f __attribute__((ext_vector_type(8)));

#define NUM_TOKENS 16384
#define HIDDEN     2048
#define NUM_EXP    64
#define TOPK       8
#define KC         64                    // K-chunk staged per double-buffer step
#define NCHUNK     (HIDDEN / KC)         // 32
#define LSTR       68                    // KC + 4 pad -> conflict-free ds_load_b64 frags
#define THREADS    128                   // 4 waves: one 16-expert tile each

// CDNA5 async global->LDS copy (no VGPR payload, tracked by ASYNCcnt).
// SADDR form: 64-bit scalar base + 32-bit per-lane byte offset.
__device__ __forceinline__ void async_copy_b128(const float* sbase, unsigned voff_bytes,
                                                void* lds_dst) {
    const unsigned lds_off = (unsigned)(size_t)lds_dst;   // addr[31:0] == LDS offset
    asm volatile("global_load_async_to_lds_b128 %0, %1, %2"
                 :
                 : "v"(lds_off), "v"(voff_bytes), "s"(sbase)
                 : "memory");
}

__device__ __forceinline__ void wait_async_done() {
#if __has_builtin(__builtin_amdgcn_s_wait_asynccnt)
    __builtin_amdgcn_s_wait_asynccnt(0);
#else
    asm volatile("s_wait_asynccnt 0x0" ::: "memory");
#endif
}

__global__ __launch_bounds__(THREADS)
void moe_router_wmma_f32(const float* __restrict__ hidden,
                         const float* __restrict__ weight,
                         float* __restrict__ w_out,
                         int*   __restrict__ e_out)
{
    // Double-buffered staged operands + block-level logits tile.
    __shared__ __align__(16) float lds_a[2][16 * LSTR];        //  8.5 KB
    __shared__ __align__(16) float lds_b[2][NUM_EXP * LSTR];   // 34.8 KB
    __shared__ __align__(16) float lds_logits[16][LSTR];       //  4.3 KB

    const int tid  = threadIdx.x;
    const int lane = tid & 31;
    const int wave = tid >> 5;                 // expert-tile id (0..3)
    const int token_base = blockIdx.x * 16;

    const int half = lane >> 4;                // 0: lanes 0-15, 1: lanes 16-31
    const int l16  = lane & 15;

    // ---- cooperative staging map: segment id -> (row = id>>4, 4-float col = id&15)
    // A chunk: 16 rows x KC floats = 256 x b128  -> 2 segs/thread
    // B chunk: 64 rows x KC floats = 1024 x b128 -> 8 segs/thread
    unsigned offA[2]; int ldsA[2];
#pragma unroll
    for (int j = 0; j < 2; ++j) {
        const int id = tid + THREADS * j;
        const int r = id >> 4, s = id & 15;
        offA[j] = (unsigned)(((token_base + r) * HIDDEN + s * 4) * 4);  // bytes
        ldsA[j] = r * LSTR + s * 4;
    }
    unsigned offB[8]; int ldsB[8];
#pragma unroll
    for (int j = 0; j < 8; ++j) {
        const int id = tid + THREADS * j;
        const int r = id >> 4, s = id & 15;
        offB[j] = (unsigned)((r * HIDDEN + s * 4) * 4);                 // bytes
        ldsB[j] = r * LSTR + s * 4;
    }

    // Stage one K-chunk (global -> LDS, fully coalesced 16B segments per lane).
    auto stage_chunk = [&](int p, int k0) {
        const unsigned kb = (unsigned)(k0 * 4);
#pragma unroll
        for (int j = 0; j < 2; ++j)
            async_copy_b128(hidden, offA[j] + kb, &lds_a[p][ldsA[j]]);
#pragma unroll
        for (int j = 0; j < 8; ++j)
            async_copy_b128(weight, offB[j] + kb, &lds_b[p][ldsB[j]]);
    };

    v8f acc = {};               // 16 tokens x 16 experts (this wave's tile)

    stage_chunk(0, 0);
    wait_async_done();
    __syncthreads();

    // Fragment base offsets inside a staged chunk (A: 16x4 f32, B: 4x16 f32).
    // Bank math: row*LSTR mod 64 = 4*row -> lanes hit banks {4*l16 + c .. +3}: all 64.
    const int arow = l16 * LSTR + 2 * half;
    const int brow = (wave * 16 + l16) * LSTR + 2 * half;

    for (int c = 0; c < NCHUNK; ++c) {
        const int p = c & 1;
        if (c + 1 < NCHUNK) {
            stage_chunk((c + 1) & 1, (c + 1) * KC);   // async DMA overlaps compute
            if (c + 2 < NCHUNK) {                     // prefetch A stream 2 chunks out
                __builtin_prefetch((const char*)hidden + offA[0] + (c + 2) * KC * 4, 0, 0);
                __builtin_prefetch((const char*)hidden + offA[1] + (c + 2) * KC * 4, 0, 0);
            }
        }

        const float* As = &lds_a[p][arow];
        const float* Bs = &lds_b[p][brow];
#pragma unroll
        for (int k = 0; k < KC; k += 4) {
            v2f a = *(const v2f*)(As + k);            // ds_load_b64, conflict-free
            v2f b = *(const v2f*)(Bs + k);
            acc = __builtin_amdgcn_wmma_f32_16x16x4_f32(
                      false, a, false, b, (short)0, acc, false, false);
        }

        wait_async_done();      // this wave's stage of chunk c+1 landed in LDS
        __syncthreads();        // all waves staged c+1 and finished reading buf p
    }

    // C/D layout: VGPR v, lanes 0-15 -> (m=v, n=l16); lanes 16-31 -> (m=v+8, n=l16)
#pragma unroll
    for (int v = 0; v < 8; ++v)
        lds_logits[v + 8 * half][wave * 16 + l16] = acc[v];
    __syncthreads();

    // One thread per token: top-8 (sorted desc, stable ties), softmax over top-8.
    if (tid < 16) {
        const float* row = &lds_logits[tid][0];

        float vals[TOPK];
        int   idx [TOPK];
#pragma unroll
        for (int i = 0; i < TOPK; ++i) { vals[i] = -3.402823466e+38f; idx[i] = 0; }

        for (int e = 0; e < NUM_EXP; ++e) {
            const float x = row[e];
            if (x <= vals[TOPK - 1]) continue;
            int pcount = 0;
#pragma unroll
            for (int i = 0; i < TOPK; ++i) pcount += (vals[i] >= x) ? 1 : 0;
#pragma unroll
            for (int j = TOPK - 1; j >= 1; --j) {
                const float pv = vals[j - 1];
                const int   pi = idx [j - 1];
                const bool shift = (j > pcount);
                const bool place = (j == pcount);
                vals[j] = shift ? pv : (place ? x : vals[j]);
                idx [j] = shift ? pi : (place ? e : idx [j]);
            }
            vals[0] = (pcount == 0) ? x : vals[0];
            idx [0] = (pcount == 0) ? e : idx [0];
        }

        const float mx = vals[0];
        float ex[TOPK];
        float s = 0.0f;
#pragma unroll
        for (int i = 0; i < TOPK; ++i) { ex[i] = __expf(vals[i] - mx); s += ex[i]; }
        const float inv = 1.0f / s;

        const int token = token_base + tid;
#pragma unroll
        for (int i = 0; i < TOPK; ++i) {
            w_out[(size_t)token * TOPK + i] = ex[i] * inv;
            e_out[(size_t)token * TOPK + i] = idx[i];
        }
    }
}

extern "C" void kernel_launch(void* const* d_in, const int* in_sizes, int n_in,
                              void* d_out, int out_size, void* d_ws, size_t ws_size,
                              hipStream_t stream) {
    (void)in_sizes; (void)n_in; (void)out_size; (void)d_ws; (void)ws_size;
    const float* hidden = (const float*)d_in[0];
    const float* weight = (const float*)d_in[1];
    float* w_out = (float*)d_out;                                  // [16384, 8] f32
    int*   e_out = (int*)d_out + (size_t)NUM_TOKENS * TOPK;        // [16384, 8] i32

    const int blocks = NUM_TOKENS / 16;     // 1024 blocks, 4 waves each
    moe_router_wmma_f32<<<blocks, THREADS, 0, stream>>>(hidden, weight, w_out, e_out);
}